// GAT_23450521436404
// MI455X (gfx1250) — compile-verified
//
#include <hip/hip_runtime.h>
#include <hip/hip_bf16.h>

#define HC 64            // heads*channels per layer output (always 64 here)
#define NEG_SLOPE 0.2f

typedef __attribute__((ext_vector_type(2))) float v2f;
typedef __attribute__((ext_vector_type(8))) float v8f;

// ---- monotone float <-> uint encoding so unsigned atomicMax == float max ----
__device__ __forceinline__ unsigned f2ord(float f) {
    unsigned b = __float_as_uint(f);
    return (b & 0x80000000u) ? ~b : (b | 0x80000000u);
}
__device__ __forceinline__ float ord2f(unsigned u) {
    unsigned b = (u & 0x80000000u) ? (u & 0x7FFFFFFFu) : ~u;
    return __uint_as_float(b);
}
#define ORD_NEG_INF 0x007FFFFFu   // f2ord(-inf)

// =====================================================================
// GEMM: C[M,64] = A[M,K] x B[K,64], f32 via V_WMMA_F32_16X16X4_F32.
// One 16x16 output tile per wave. M is a multiple of 16 (50000 = 3125*16).
// A 16x4 lane layout: lane&15 = row M; lanes>=16 hold K+2,K+3.
// B 4x16 lane layout: lane&15 = col N; lanes>=16 hold K+2,K+3.
// C/D:              : lane&15 = col N; vgpr v -> row v + 8*(lane>>4).
// =====================================================================
template <int K>
__global__ void k_gemm_wmma(const float* __restrict__ A,
                            const float* __restrict__ B,
                            float* __restrict__ C, int mtiles) {
    const int lane = threadIdx.x & 31;
    const int wave = threadIdx.x >> 5;
    const int ntiles = HC / 16;                         // 4
    const int tile = blockIdx.x * (blockDim.x >> 5) + wave;  // wave-uniform
    if (tile >= mtiles * ntiles) return;                // whole-wave exit
    const int mt   = tile / ntiles;
    const int nt   = tile % ntiles;
    const int half = lane >> 4;       // 0 or 1
    const int l15  = lane & 15;
    const int rowA = mt * 16 + l15;
    const int kb   = half * 2;        // K sub-offset held by this lane
    const int colB = nt * 16 + l15;

    v8f acc = {};
#pragma unroll
    for (int k = 0; k < K; k += 4) {
        v2f a, b;
        a.x = A[rowA * K + k + kb];
        a.y = A[rowA * K + k + kb + 1];
        b.x = B[(k + kb) * HC + colB];
        b.y = B[(k + kb + 1) * HC + colB];
        acc = __builtin_amdgcn_wmma_f32_16x16x4_f32(
            false, a, false, b, (short)0, acc, false, false);
    }
#pragma unroll
    for (int v = 0; v < 8; ++v) {
        const int row = mt * 16 + v + half * 8;
        C[row * HC + nt * 16 + l15] = acc[v];
    }
}

// ---- per-node attention coefficients: al_s/al_d [N,H] = sum_c h*a ----
__global__ void k_coef(const float* __restrict__ h,
                       const float* __restrict__ a_src,
                       const float* __restrict__ a_dst,
                       float* __restrict__ al_s, float* __restrict__ al_d,
                       int n, int H, int C) {
    int idx = blockIdx.x * blockDim.x + threadIdx.x;
    if (idx >= n * H) return;
    int node = idx / H, head = idx % H;
    const float* hp = h + (size_t)node * HC + head * C;
    const float* as = a_src + head * C;
    const float* ad = a_dst + head * C;
    float ss = 0.f, sd = 0.f;
    for (int c = 0; c < C; ++c) {
        float v = hp[c];
        ss += v * as[c];
        sd += v * ad[c];
    }
    al_s[idx] = ss;
    al_d[idx] = sd;
}

// ---- zero agg, init segment max/sum ----
__global__ void k_init(float* __restrict__ agg, unsigned* __restrict__ mbits,
                       float* __restrict__ ssum, int n, int H) {
    int idx = blockIdx.x * blockDim.x + threadIdx.x;
    if (idx < n * HC) agg[idx] = 0.f;
    if (idx < n * H) { mbits[idx] = ORD_NEG_INF; ssum[idx] = 0.f; }
}

// ---- edge pass 1: leaky-relu logit + segment max (atomic) ----
__global__ void k_edge_logit(const int* __restrict__ esrc, const int* __restrict__ edst,
                             const float* __restrict__ al_s, const float* __restrict__ al_d,
                             float* __restrict__ elog, unsigned* __restrict__ mbits,
                             int e_orig, int e_tot, int H) {
    int idx = blockIdx.x * blockDim.x + threadIdx.x;
    if (idx >= e_tot * H) return;
    int e = idx / H, hd = idx % H;
    int s, d;
    if (e < e_orig) { s = esrc[e]; d = edst[e]; } else { s = d = e - e_orig; }
    float x  = al_s[s * H + hd] + al_d[d * H + hd];
    float lg = x >= 0.f ? x : NEG_SLOPE * x;
    elog[idx] = lg;
    atomicMax(&mbits[d * H + hd], f2ord(lg));
}

// ---- edge pass 2: e = exp(l - m), segment sum (atomic) ----
__global__ void k_edge_exp(const int* __restrict__ edst,
                           float* __restrict__ elog,
                           const unsigned* __restrict__ mbits,
                           float* __restrict__ ssum,
                           int e_orig, int e_tot, int H) {
    int idx = blockIdx.x * blockDim.x + threadIdx.x;
    if (idx >= e_tot * H) return;
    int e = idx / H, hd = idx % H;
    int d  = (e < e_orig) ? edst[e] : e - e_orig;
    float m  = ord2f(mbits[d * H + hd]);
    float ex = expf(elog[idx] - m);
    elog[idx] = ex;
    atomicAdd(&ssum[d * H + hd], ex);
}

// ---- edge pass 3: scatter h[src]*alpha into agg[dst]; 8 threads/edge ----
__global__ void k_edge_scatter(const int* __restrict__ esrc, const int* __restrict__ edst,
                               const float* __restrict__ elog, const float* __restrict__ ssum,
                               const float* __restrict__ hlin, float* __restrict__ agg,
                               int e_orig, int e_tot, int H) {
    int idx = blockIdx.x * blockDim.x + threadIdx.x;
    if (idx >= e_tot * 8) return;
    int e = idx >> 3, part = idx & 7;   // part selects 8 contiguous channels
    int s, d;
    if (e < e_orig) { s = esrc[e]; d = edst[e]; } else { s = d = e - e_orig; }
    int aidx = (H == 8) ? part : 0;
    float alpha = elog[e * H + aidx] / (ssum[d * H + aidx] + 1e-16f);
    const float* hp = hlin + (size_t)s * HC + part * 8;
    float* ap = agg + (size_t)d * HC + part * 8;
#pragma unroll
    for (int c = 0; c < 8; ++c) atomicAdd(&ap[c], hp[c] * alpha);
}

// ---- finalize: out = agg + bias, optional ELU ----
__global__ void k_finalize(const float* __restrict__ agg, const float* __restrict__ bias,
                           float* __restrict__ out, int n, int do_elu) {
    int idx = blockIdx.x * blockDim.x + threadIdx.x;
    if (idx >= n * HC) return;
    float v = agg[idx] + bias[idx % HC];
    if (do_elu) v = v > 0.f ? v : expf(v) - 1.f;
    out[idx] = v;
}

extern "C" void kernel_launch(void* const* d_in, const int* in_sizes, int n_in,
                              void* d_out, int out_size, void* d_ws, size_t ws_size,
                              hipStream_t stream) {
    const float* x    = (const float*)d_in[0];
    const int*   eidx = (const int*)d_in[1];
    const float* W1   = (const float*)d_in[2];
    const float* as1  = (const float*)d_in[3];
    const float* ad1  = (const float*)d_in[4];
    const float* b1   = (const float*)d_in[5];
    const float* W2   = (const float*)d_in[6];
    const float* as2  = (const float*)d_in[7];
    const float* ad2  = (const float*)d_in[8];
    const float* b2   = (const float*)d_in[9];
    const float* W3   = (const float*)d_in[10];
    const float* as3  = (const float*)d_in[11];
    const float* ad3  = (const float*)d_in[12];
    const float* b3   = (const float*)d_in[13];

    const int N     = in_sizes[0] / 128;   // 50000
    const int E     = in_sizes[1] / 2;     // 800000
    const int E_tot = E + N;               // + self loops
    const int* esrc = eidx;
    const int* edst = eidx + E;

    // workspace layout (floats)
    float* ws   = (float*)d_ws;
    size_t off  = 0;
    float* bufA = ws + off; off += (size_t)N * HC;   // h_lin
    float* bufB = ws + off; off += (size_t)N * HC;   // activated features
    float* agg  = ws + off; off += (size_t)N * HC;
    float* al_s = ws + off; off += (size_t)N * 8;
    float* al_d = ws + off; off += (size_t)N * 8;
    unsigned* mbits = (unsigned*)(ws + off); off += (size_t)N * 8;
    float* ssum = ws + off; off += (size_t)N * 8;
    float* elog = ws + off; off += (size_t)E_tot * 8;

    const int TPB = 256;
    const int mtiles = N / 16;                 // 3125, exact
    const int tiles  = mtiles * (HC / 16);     // 12500 waves
    const int gemm_blocks = (tiles + (TPB / 32) - 1) / (TPB / 32);

    auto layer = [&](const float* hin, int K, const float* W,
                     const float* asrc, const float* adst, const float* bias,
                     int H, int C, float* hlin, float* hout, int do_elu) {
        if (K == 128)
            k_gemm_wmma<128><<<gemm_blocks, TPB, 0, stream>>>(hin, W, hlin, mtiles);
        else
            k_gemm_wmma<64><<<gemm_blocks, TPB, 0, stream>>>(hin, W, hlin, mtiles);
        k_coef<<<(N * H + TPB - 1) / TPB, TPB, 0, stream>>>(hlin, asrc, adst,
                                                            al_s, al_d, N, H, C);
        k_init<<<(N * HC + TPB - 1) / TPB, TPB, 0, stream>>>(agg, mbits, ssum, N, H);
        const int eh = E_tot * H;
        k_edge_logit<<<(eh + TPB - 1) / TPB, TPB, 0, stream>>>(
            esrc, edst, al_s, al_d, elog, mbits, E, E_tot, H);
        k_edge_exp<<<(eh + TPB - 1) / TPB, TPB, 0, stream>>>(
            edst, elog, mbits, ssum, E, E_tot, H);
        const int e8 = E_tot * 8;
        k_edge_scatter<<<(e8 + TPB - 1) / TPB, TPB, 0, stream>>>(
            esrc, edst, elog, ssum, hlin, agg, E, E_tot, H);
        k_finalize<<<(N * HC + TPB - 1) / TPB, TPB, 0, stream>>>(agg, bias, hout, N, do_elu);
    };

    // Layer 1: x[N,128] -> bufB (ELU)
    layer(x,    128, W1, as1, ad1, b1, 8, 8,  bufA, bufB, 1);
    // Layer 2: bufB -> bufB (ELU); gemm writes bufA, finalize overwrites bufB
    layer(bufB,  64, W2, as2, ad2, b2, 8, 8,  bufA, bufB, 1);
    // Layer 3: bufB -> d_out (1 head, 64 ch, no ELU, mean over 1 head = identity)
    layer(bufB,  64, W3, as3, ad3, b3, 1, 64, bufA, (float*)d_out, 0);
}